// message_79869211837100
// MI455X (gfx1250) — compile-verified
//
#include <hip/hip_runtime.h>

#define N_EDGES_C 320000
#define N_NODES_C 20000
#define FDIM 128
#define N_RBF_C 20
#define R_CUT_F 5.0f
#define PI_F 3.14159265358979f

// Column-major LDS layout for WMMA B fragments: W[k][n] -> lds[n*STRIDE + k].
// Per-lane base = (n0+m)*STRIDE, K-step offsets k*4 <= 508 B fold into the DS
// 16-bit immediate. STRIDE ≡ 4 (mod 64) dwords -> the 64 dwords touched by one
// ds_load_b64 (two half-waves at k and k+2) land on 64 distinct banks.
#define WPHI_STRIDE 132   // >= 128, 132 mod 64 == 4
#define WW_STRIDE   28    // >= 20,  28*m mod 64 enumerates multiples of 4

typedef float v2f __attribute__((ext_vector_type(2)));
typedef float v4f __attribute__((ext_vector_type(4)));
typedef float v8f __attribute__((ext_vector_type(8)));

// ---------------------------------------------------------------------------
// Kernel 0: zero the output (atomic accumulation target) and norm accumulator
// ---------------------------------------------------------------------------
__global__ void init_zero(float* __restrict__ out, int n, float* __restrict__ ws) {
    int stride = gridDim.x * blockDim.x;
    for (int i = blockIdx.x * blockDim.x + threadIdx.x; i < n; i += stride)
        out[i] = 0.0f;
    if (blockIdx.x == 0 && threadIdx.x == 0)
        ws[0] = 0.0f;
}

// ---------------------------------------------------------------------------
// Kernel 1: global Frobenius norm^2 of r (reference's jnp.linalg.norm has no
// axis -> one scalar over the whole (E,3) array)
// ---------------------------------------------------------------------------
__global__ void reduce_r2(const float* __restrict__ r, float* __restrict__ acc) {
    const int n = N_EDGES_C * 3;
    float sum = 0.0f;
    int stride = gridDim.x * blockDim.x;
    for (int i = blockIdx.x * blockDim.x + threadIdx.x; i < n; i += stride) {
        float x = r[i];
        sum += x * x;
    }
    #pragma unroll
    for (int off = 16; off > 0; off >>= 1)
        sum += __shfl_down(sum, off, 32);
    if ((threadIdx.x & 31) == 0)
        unsafeAtomicAdd(acc, sum);
}

// ---------------------------------------------------------------------------
// Kernel 2: main message kernel. 256 threads = 8 wave32; one wave owns 16
// edges. Weights are staged once per workgroup into LDS (column-major) so
// every WMMA B fragment is one conflict-free ds_load_b64 with an immediate
// offset. Biases staged via GLOBAL_LOAD_ASYNC_TO_LDS_B128 (ASYNCcnt path).
//
// fp32 16x16x4 WMMA register layouts (ISA 7.12.2):
//   A (16x4):  lane = half*16+m ; VGPR0 = A[m][k0+2*half], VGPR1 = A[m][k0+2*half+1]
//   B (4x16):  VGPR0 = B[k0+2*half][n0+m], VGPR1 = B[k0+2*half+1][n0+m]
//   C/D (16x16): VGPR j = C[row=j+8*half][col=n0+m]
// ---------------------------------------------------------------------------
__global__ __launch_bounds__(256) void painn_message(
    const float* __restrict__ s,      // E x 128
    const float* __restrict__ r,      // E x 3
    const float* __restrict__ v,      // E x 3 x 128
    const int*   __restrict__ idx_i,  // E
    const float* __restrict__ W_phi,  // 128 x 384
    const float* __restrict__ b_phi,  // 384
    const float* __restrict__ W_w,    // 20 x 384
    const float* __restrict__ b_w,    // 384
    const float* __restrict__ norm_ws,
    float* __restrict__ out_v,        // NODES x 3 x 128
    float* __restrict__ out_s)        // NODES x 128
{
    __shared__ __attribute__((aligned(16))) float lds_wphi[384 * WPHI_STRIDE]; // 202752 B
    __shared__ __attribute__((aligned(16))) float lds_ww[384 * WW_STRIDE];     //  43008 B
    __shared__ __attribute__((aligned(16))) float lds_bias[2 * 384];           //   3072 B

    const int tid  = threadIdx.x;
    const int lane = tid & 31;
    const int wave = tid >> 5;

    // ---- stage biases (async DMA) + weights (transpose to col-major) -------
    {
        for (int c = tid; c < 96; c += 256) {
            unsigned lds_off = (unsigned)(uintptr_t)(lds_bias + c * 4);
            unsigned long long gaddr = (unsigned long long)(uintptr_t)(b_phi + c * 4);
            asm volatile("global_load_async_to_lds_b128 %0, %1, off"
                         :: "v"(lds_off), "v"(gaddr) : "memory");
        }
        for (int c = tid; c < 96; c += 256) {
            unsigned lds_off = (unsigned)(uintptr_t)(lds_bias + 384 + c * 4);
            unsigned long long gaddr = (unsigned long long)(uintptr_t)(b_w + c * 4);
            asm volatile("global_load_async_to_lds_b128 %0, %1, off"
                         :: "v"(lds_off), "v"(gaddr) : "memory");
        }

        // W_phi: 128 rows x 384 cols -> lds_wphi[n*132 + k]
        for (int c = tid; c < FDIM * 96; c += 256) {
            int k    = c / 96;
            int col4 = (c % 96) * 4;
            v4f w = *(const v4f*)(W_phi + (size_t)k * 384 + col4);
            float* dst = lds_wphi + col4 * WPHI_STRIDE + k;
            dst[0 * WPHI_STRIDE] = w.x;
            dst[1 * WPHI_STRIDE] = w.y;
            dst[2 * WPHI_STRIDE] = w.z;
            dst[3 * WPHI_STRIDE] = w.w;
        }
        // W_w: 20 rows x 384 cols -> lds_ww[n*28 + k]
        for (int c = tid; c < N_RBF_C * 96; c += 256) {
            int k    = c / 96;
            int col4 = (c % 96) * 4;
            v4f w = *(const v4f*)(W_w + (size_t)k * 384 + col4);
            float* dst = lds_ww + col4 * WW_STRIDE + k;
            dst[0 * WW_STRIDE] = w.x;
            dst[1 * WW_STRIDE] = w.y;
            dst[2 * WW_STRIDE] = w.z;
            dst[3 * WW_STRIDE] = w.w;
        }
        asm volatile("s_wait_asynccnt 0x0" ::: "memory");
        __syncthreads();
    }

    const int gw = blockIdx.x * 8 + wave;     // 8 waves per 256-thread block
    const int e0 = gw * 16;                   // first edge of this wave's tile
    if (e0 >= N_EDGES_C) return;              // wave-uniform

    const int half = lane >> 4;               // 0 or 1
    const int m    = lane & 15;               // row (A) / column (B,C,D) index

    const float inv_gnorm = rsqrtf(norm_ws[0]);   // 1 / ||r||_F (global)

    // ---- cache A tile of s: 32 K-steps x v2f/lane (b64 loads) --------------
    v2f As[32];
    {
        const float* srow = s + (size_t)(e0 + m) * FDIM;
        #pragma unroll
        for (int ks = 0; ks < 32; ++ks) {
            int k = ks * 4 + 2 * half;        // even -> 8B aligned
            As[ks] = *(const v2f*)(srow + k);
        }
    }

    // ---- build RBF A tile (16 x 20): 5 K-steps x v2f/lane ------------------
    v2f Ar[5];
    {
        const float* rrow = r + (size_t)(e0 + m) * 3;
        float rx = rrow[0], ry = rrow[1], rz = rrow[2];
        float rn = sqrtf(rx * rx + ry * ry + rz * rz);
        float inv_rn = 1.0f / rn;
        #pragma unroll
        for (int ks = 0; ks < 5; ++ks) {
            #pragma unroll
            for (int t = 0; t < 2; ++t) {
                int k = ks * 4 + 2 * half + t;          // 0..19
                float nval = (float)(k + 1);
                float raw  = __sinf(nval * (PI_F / R_CUT_F) * rn) * inv_rn;
                float val  = (raw <= R_CUT_F)
                               ? 0.5f * (__cosf(PI_F * raw / R_CUT_F) + 1.0f)
                               : 0.0f;
                Ar[ks][t] = val;
            }
        }
    }

    // ---- per-lane edge metadata for the 8 output rows this lane owns -------
    int   idxj[8];
    float rdir[8][3];
    #pragma unroll
    for (int j = 0; j < 8; ++j) {
        int e = e0 + j + 8 * half;
        idxj[j] = idx_i[e];
        rdir[j][0] = r[(size_t)e * 3 + 0] * inv_gnorm;
        rdir[j][1] = r[(size_t)e * 3 + 1] * inv_gnorm;
        rdir[j][2] = r[(size_t)e * 3 + 2] * inv_gnorm;
    }

    const int koff = 2 * half;               // K offset of this half-wave

    // ---- 8 feature tiles of 16 columns each --------------------------------
    #pragma unroll 1
    for (int ft = 0; ft < 8; ++ft) {
        v8f S[3];
        #pragma unroll
        for (int part = 0; part < 3; ++part) {
            const int n0 = part * FDIM + ft * 16;

            // phi GEMM: K = 128 in 32 steps of 4; B = ds_load_b64.
            // Opaque base index: stops reassociation into base+huge-constant,
            // so every K-step folds into the DS 16-bit immediate.
            int cbP = (n0 + m) * WPHI_STRIDE + koff;
            asm volatile("" : "+v"(cbP));
            const float* colP = lds_wphi + cbP;
            v8f accP = {0.f, 0.f, 0.f, 0.f, 0.f, 0.f, 0.f, 0.f};
            #pragma unroll
            for (int ks = 0; ks < 32; ++ks) {
                v2f b = *(const v2f*)(colP + ks * 4);   // imm offset <= 508 B
                accP = __builtin_amdgcn_wmma_f32_16x16x4_f32(
                           false, As[ks], false, b, (short)0, accP, false, false);
            }
            const float bp = lds_bias[n0 + m];

            // w GEMM: K = 20 in 5 steps of 4
            int cbW = (n0 + m) * WW_STRIDE + koff;
            asm volatile("" : "+v"(cbW));
            const float* colW = lds_ww + cbW;
            v8f accW = {0.f, 0.f, 0.f, 0.f, 0.f, 0.f, 0.f, 0.f};
            #pragma unroll
            for (int ks = 0; ks < 5; ++ks) {
                v2f b = *(const v2f*)(colW + ks * 4);   // imm offset <= 72 B
                accW = __builtin_amdgcn_wmma_f32_16x16x4_f32(
                           false, Ar[ks], false, b, (short)0, accW, false, false);
            }
            const float bw = lds_bias[384 + n0 + m];

            v8f sp;
            #pragma unroll
            for (int j = 0; j < 8; ++j)
                sp[j] = (accP[j] + bp) * (accW[j] + bw);
            S[part] = sp;
        }

        const int f = ft * 16 + m;   // feature column this lane owns

        // out_s += sp1
        #pragma unroll
        for (int j = 0; j < 8; ++j)
            unsafeAtomicAdd(out_s + (size_t)idxj[j] * FDIM + f, S[1][j]);

        // out_v += sp2 * dir + sp0 * v
        #pragma unroll
        for (int j = 0; j < 8; ++j) {
            int e = e0 + j + 8 * half;
            const float* vrow = v + (size_t)e * (3 * FDIM) + f;
            float* orow = out_v + (size_t)idxj[j] * (3 * FDIM) + f;
            #pragma unroll
            for (int d = 0; d < 3; ++d) {
                float msg = S[2][j] * rdir[j][d] + S[0][j] * vrow[(size_t)d * FDIM];
                unsafeAtomicAdd(orow + (size_t)d * FDIM, msg);
            }
        }
    }
}

// ---------------------------------------------------------------------------
extern "C" void kernel_launch(void* const* d_in, const int* in_sizes, int n_in,
                              void* d_out, int out_size, void* d_ws, size_t ws_size,
                              hipStream_t stream) {
    const float* s     = (const float*)d_in[0];
    const float* r     = (const float*)d_in[1];
    const float* v     = (const float*)d_in[2];
    const int*   idx_i = (const int*)d_in[3];
    const float* W_phi = (const float*)d_in[4];
    const float* b_phi = (const float*)d_in[5];
    const float* W_w   = (const float*)d_in[6];
    const float* b_w   = (const float*)d_in[7];

    float* out   = (float*)d_out;
    float* out_v = out;                                   // NODES*3*128
    float* out_s = out + (size_t)N_NODES_C * 3 * FDIM;    // NODES*128
    float* ws    = (float*)d_ws;                          // [0] = sum(r^2)

    init_zero<<<512, 256, 0, stream>>>(out, out_size, ws);
    reduce_r2<<<512, 256, 0, stream>>>(r, ws);

    // 320000 edges / 16 per wave / 8 waves per block = 2500 blocks exactly
    painn_message<<<2500, 256, 0, stream>>>(s, r, v, idx_i, W_phi, b_phi,
                                            W_w, b_w, ws, out_v, out_s);
}